// LocalNormalizedCutLoss_38706245272152
// MI455X (gfx1250) — compile-verified
//
#include <hip/hip_runtime.h>
#include <hip/hip_bf16.h>

// Problem constants (match reference shapes exactly)
#define B_   4
#define C_   21
#define H_   512
#define W_   512
#define PAD  2
#define TILE 16
#define EXT  (TILE + 2 * PAD)   // 20
#define EXT2 (EXT * EXT)        // 400
#define NTX  (W_ / TILE)        // 32
#define NTY  (H_ / TILE)        // 32
#define NBLK (B_ * NTX * NTY)   // 4096
#define INV2SIG2 200.0f         // 1/(2*0.05^2)

#if defined(__AMDGCN__) && __has_builtin(__builtin_amdgcn_global_load_async_to_lds_b32)
#define HAVE_ASYNC 1
#else
#define HAVE_ASYNC 0
#endif

#define AS1 __attribute__((address_space(1)))
#define AS3 __attribute__((address_space(3)))

// One float: global -> LDS. Async (CDNA5 GLOBAL_LOAD_ASYNC_TO_LDS_B32) when available.
__device__ __forceinline__ void g2l_f32(float* lds_dst, const float* g_src) {
#if HAVE_ASYNC
    __builtin_amdgcn_global_load_async_to_lds_b32(
        (AS1 int*)g_src, (AS3 int*)lds_dst, /*offset=*/0, /*cpol=*/0);
#else
    *lds_dst = *g_src;
#endif
}

__device__ __forceinline__ void wait_async_all() {
#if HAVE_ASYNC
#if __has_builtin(__builtin_amdgcn_s_wait_asynccnt)
    __builtin_amdgcn_s_wait_asynccnt(0);
#else
    asm volatile("s_wait_asynccnt 0" ::: "memory");
#endif
#endif
}

// reflect padding (jnp.pad mode='reflect'): -1->1, -2->2, H->H-2, H+1->H-3
__device__ __forceinline__ int reflect_idx(int i, int n) {
    if (i < 0) i = -i;
    if (i >= n) i = 2 * n - 2 - i;
    return i;
}

__global__ __launch_bounds__(256)
void lncl_tile_kernel(const float* __restrict__ preds,
                      const float* __restrict__ images,
                      float* __restrict__ partial) {
    __shared__ float sP[C_ * EXT2];   // probs, channel-major: [c][p]
    __shared__ float sI[3 * EXT2];    // image halo, channel-major
    __shared__ float sred[256];

    const int tid = threadIdx.x;
    const int bx = blockIdx.x, by = blockIdx.y, b = blockIdx.z;
    const int x0 = bx * TILE, y0 = by * TILE;

    // ---- Phase 1: gather halo tile (reflect-padded) into LDS ----
    // preds: 21 channels * 400 ext pixels
    for (int idx = tid; idx < C_ * EXT2; idx += 256) {
        int c = idx / EXT2;
        int p = idx - c * EXT2;
        int ey = p / EXT;
        int ex = p - ey * EXT;
        int gy = reflect_idx(y0 + ey - PAD, H_);
        int gx = reflect_idx(x0 + ex - PAD, W_);
        const float* src = preds + (((size_t)b * C_ + c) * H_ + gy) * W_ + gx;
        g2l_f32(&sP[idx], src);
    }
    // images: 3 channels * 400 ext pixels
    for (int idx = tid; idx < 3 * EXT2; idx += 256) {
        int c = idx / EXT2;
        int p = idx - c * EXT2;
        int ey = p / EXT;
        int ex = p - ey * EXT;
        int gy = reflect_idx(y0 + ey - PAD, H_);
        int gx = reflect_idx(x0 + ex - PAD, W_);
        const float* src = images + (((size_t)b * 3 + c) * H_ + gy) * W_ + gx;
        g2l_f32(&sI[idx], src);
    }
    wait_async_all();
    __syncthreads();

    // ---- Phase 2: in-place softmax over C for every ext pixel ----
    for (int p = tid; p < EXT2; p += 256) {
        float m = -3.402823466e+38f;
        #pragma unroll
        for (int c = 0; c < C_; ++c) m = fmaxf(m, sP[c * EXT2 + p]);
        float s = 0.0f;
        #pragma unroll
        for (int c = 0; c < C_; ++c) {
            float e = __expf(sP[c * EXT2 + p] - m);
            sP[c * EXT2 + p] = e;
            s += e;
        }
        float r = __frcp_rn(s);
        #pragma unroll
        for (int c = 0; c < C_; ++c) sP[c * EXT2 + p] *= r;
    }
    __syncthreads();

    // ---- Phase 3: 24-offset loss for this thread's pixel ----
    const int tx = tid & (TILE - 1);
    const int ty = tid >> 4;
    const int ep = (ty + PAD) * EXT + (tx + PAD);

    float pc[C_];
    #pragma unroll
    for (int c = 0; c < C_; ++c) pc[c] = sP[c * EXT2 + ep];
    const float r0 = sI[0 * EXT2 + ep];
    const float g0 = sI[1 * EXT2 + ep];
    const float b0 = sI[2 * EXT2 + ep];

    float acc = 0.0f;
    #pragma unroll
    for (int dy = -PAD; dy <= PAD; ++dy) {
        #pragma unroll
        for (int dx = -PAD; dx <= PAD; ++dx) {
            if (dy == 0 && dx == 0) continue;
            const int q = ep + dy * EXT + dx;
            float dr = r0 - sI[0 * EXT2 + q];
            float dg = g0 - sI[1 * EXT2 + q];
            float db = b0 - sI[2 * EXT2 + q];
            float cd = fmaf(dr, dr, fmaf(dg, dg, db * db));
            float aff = __expf(-cd * INV2SIG2);
            float s = 0.0f;
            #pragma unroll
            for (int c = 0; c < C_; ++c) {
                float t = pc[c] - sP[c * EXT2 + q];
                s = fmaf(t, t, s);
            }
            acc = fmaf(aff, s, acc);
        }
    }

    // ---- Block reduction -> per-block partial (deterministic) ----
    sred[tid] = acc;
    __syncthreads();
    #pragma unroll
    for (int k = 128; k > 0; k >>= 1) {
        if (tid < k) sred[tid] += sred[tid + k];
        __syncthreads();
    }
    if (tid == 0) {
        partial[(b * NTY + by) * NTX + bx] = sred[0];
    }
}

// Fixed-order final reduction of 4096 block partials -> scalar loss
__global__ __launch_bounds__(256)
void lncl_reduce_kernel(const float* __restrict__ partial, float* __restrict__ out) {
    __shared__ float sred[256];
    const int tid = threadIdx.x;
    float s = 0.0f;
    for (int i = tid; i < NBLK; i += 256) s += partial[i];
    sred[tid] = s;
    __syncthreads();
    #pragma unroll
    for (int k = 128; k > 0; k >>= 1) {
        if (tid < k) sred[tid] += sred[tid + k];
        __syncthreads();
    }
    if (tid == 0) {
        // loss = sum / (B*H*W) / (24*C)
        const float scale = (float)(1.0 / ((double)B_ * H_ * W_ * 24.0 * C_));
        out[0] = sred[0] * scale;
    }
}

extern "C" void kernel_launch(void* const* d_in, const int* in_sizes, int n_in,
                              void* d_out, int out_size, void* d_ws, size_t ws_size,
                              hipStream_t stream) {
    (void)in_sizes; (void)n_in; (void)out_size; (void)ws_size;
    const float* preds  = (const float*)d_in[0];
    const float* images = (const float*)d_in[1];
    float* out = (float*)d_out;
    float* ws  = (float*)d_ws;   // NBLK floats of partial sums

    dim3 grid(NTX, NTY, B_);
    lncl_tile_kernel<<<grid, 256, 0, stream>>>(preds, images, ws);
    lncl_reduce_kernel<<<1, 256, 0, stream>>>(ws, out);
}